// TFGRU_28793460752665
// MI455X (gfx1250) — compile-verified
//
#include <hip/hip_runtime.h>
#include <math.h>

typedef float v2f __attribute__((ext_vector_type(2)));
typedef float v8f __attribute__((ext_vector_type(8)));

constexpr int Bb = 4, Nn = 80, CHh = 64;
constexpr float GAM = 4.0f;     // 1 / 0.5^2
constexpr float EPSc = 1e-6f;

__device__ __forceinline__ v8f zero8() {
  v8f z;
#pragma unroll
  for (int i = 0; i < 8; ++i) z[i] = 0.0f;
  return z;
}

// A fragment: 16x4 f32 tile of W (row stride 48). Lane l<16 holds row l, K = 4s+0/1;
// lane l>=16 holds row l-16, K = 4s+2/3  (VGPR0={K0,K2}, VGPR1={K1,K3} per ISA layout).
__device__ __forceinline__ v2f ldA(const float* __restrict__ W, int row, int s, int half) {
  const float* p = W + row * 48 + 4 * s + 2 * half;
  v2f r; r.x = p[0]; r.y = p[1]; return r;
}
// B fragment: 4x16 f32 tile of G ([48][16] in LDS). Mirror layout of A.
__device__ __forceinline__ v2f ldB(const float* G, int col, int s, int half) {
  const int r0 = 4 * s + 2 * half;
  v2f r; r.x = G[r0 * 16 + col]; r.y = G[(r0 + 1) * 16 + col]; return r;
}

#define GEMM48(ACC, WPTR, GPTR)                                                          \
  _Pragma("unroll") for (int s = 0; s < 12; ++s) {                                       \
    v2f a_ = ldA(WPTR, col, s, half);                                                    \
    v2f b_ = ldB(GPTR, col, s, half);                                                    \
    ACC = __builtin_amdgcn_wmma_f32_16x16x4_f32(false, a_, false, b_, (short)0, ACC,     \
                                                false, false);                           \
  }

// One wave32 per (n,b). Computes h_conv[n, :, b, :] without materializing Kmat.
__global__ __launch_bounds__(32) void conv_kernel(
    const float* __restrict__ h_nei, const float* __restrict__ h_nei_diff,
    const float* __restrict__ mask,
    const float* __restrict__ w00, const float* __restrict__ w10,
    const float* __restrict__ w01, const float* __restrict__ w11,
    float* __restrict__ h_conv) {
  __shared__ float Gs[48 * 16], Gp[48 * 16];
  __shared__ float Gh[3][48 * 16], Gc[3][48 * 16], Gt[3][48 * 16];
  __shared__ float nhL[3][16];

  const int n = blockIdx.x / Nn, b = blockIdx.x % Nn;
  const int lane = threadIdx.x;
  const int col = lane & 15;   // = A row = B col = D col
  const int half = lane >> 4;

  for (int tile = 0; tile < 5; ++tile) {
    __syncthreads();
    const int a = tile * 16 + col;

    // ---- phase 1: per-neighbor geometry + rbf-weighted feature staging ----
    const float* dp = h_nei_diff + (((size_t)n * Nn + b) * Nn + a) * 3;
    const float dx = dp[0], dy = dp[1], dz = dp[2];
    const float rad = sqrtf(dx * dx + dy * dy + dz * dz + EPSc);
    const float inv = 1.0f / rad;
    const float nh0 = dx * inv, nh1 = dy * inv, nh2 = dz * inv;
    const float mk = mask[((size_t)n * Nn + b) * Nn + a];
    float rb[3];
    rb[0] = __expf(-GAM * rad * rad) * mk;
    float t = rad - 0.5f; rb[1] = __expf(-GAM * t * t) * mk;
    t = rad - 1.0f;       rb[2] = __expf(-GAM * t * t) * mk;
    if (half == 0) { nhL[0][col] = nh0; nhL[1][col] = nh1; nhL[2][col] = nh2; }

#pragma unroll
    for (int vv = 0; vv < 8; ++vv) {
      const int v = half * 8 + vv;
      const float hs = h_nei[((size_t)n * CHh + v) * Nn + a];
      const float h0 = h_nei[((size_t)n * CHh + 16 + 3 * v + 0) * Nn + a];
      const float h1 = h_nei[((size_t)n * CHh + 16 + 3 * v + 1) * Nn + a];
      const float h2 = h_nei[((size_t)n * CHh + 16 + 3 * v + 2) * Nn + a];
      const float P = nh0 * h0 + nh1 * h1 + nh2 * h2;            // nh . h_v
      const float c0 = h1 * nh2 - h2 * nh1;                      // (h_v x nh)
      const float c1 = h2 * nh0 - h0 * nh2;
      const float c2 = h0 * nh1 - h1 * nh0;
      const float q0 = nh0 * P - h0 * (1.0f / 3.0f);             // (nh nh^T - I/3) h_v
      const float q1 = nh1 * P - h1 * (1.0f / 3.0f);
      const float q2 = nh2 * P - h2 * (1.0f / 3.0f);
#pragma unroll
      for (int k = 0; k < 3; ++k) {
        const int p = (v * 3 + k) * 16 + col;   // inner index matches W row layout u*48 + 3v+k
        const float rk = rb[k];
        Gs[p] = rk * hs;  Gp[p] = rk * P;
        Gh[0][p] = rk * h0; Gh[1][p] = rk * h1; Gh[2][p] = rk * h2;
        Gc[0][p] = rk * c0; Gc[1][p] = rk * c1; Gc[2][p] = rk * c2;
        Gt[0][p] = rk * q0; Gt[1][p] = rk * q1; Gt[2][p] = rk * q2;
      }
    }
    __syncthreads();

    // ---- phase 2: twelve 16x48x16 WMMA GEMM tiles ----
    v8f accS = zero8();             // K00 @ h_s  +  K01 @ h_v
    GEMM48(accS, w00, Gs)
    GEMM48(accS, w01, Gp)
    v8f acc10 = zero8();            // radial part of K10
    GEMM48(acc10, w10, Gs)
    v8f accV[3];
#pragma unroll
    for (int i = 0; i < 3; ++i) {
      accV[i] = zero8();            // K11: identity + cross + traceless-outer parts
      GEMM48(accV[i], w11 + 0 * 768, Gh[i])
      GEMM48(accV[i], w11 + 1 * 768, Gc[i])
      GEMM48(accV[i], w11 + 2 * 768, Gt[i])
    }

    // ---- write h_conv[n, d, b, a] ----
#pragma unroll
    for (int r = 0; r < 8; ++r) {
      const int u = r + 8 * half;   // D layout: VGPR r -> M=r (lanes<16) / M=8+r
      h_conv[(((size_t)n * CHh + u) * Nn + b) * Nn + a] = accS[r];
      const float o10 = acc10[r];
#pragma unroll
      for (int i = 0; i < 3; ++i)
        h_conv[(((size_t)n * CHh + 16 + 3 * u + i) * Nn + b) * Nn + a] =
            o10 * nhL[i][col] + accV[i][r];
    }
  }
}

__device__ __forceinline__ float sigm(float v) { return 1.0f / (1.0f + __expf(-v)); }

// One 64-thread block per (n,b): sum_h, r-gate, sum_g, z, pre_h, new_h.
__global__ __launch_bounds__(64) void post_kernel(
    const float* __restrict__ x, const float* __restrict__ h_conv,
    const float* __restrict__ zw0, const float* __restrict__ zw1,
    const float* __restrict__ r1w0, const float* __restrict__ r1w1,
    const float* __restrict__ r2w0, const float* __restrict__ r2w1,
    const float* __restrict__ pw0, const float* __restrict__ pw1,
    float* __restrict__ out) {
  __shared__ float hc[64], xa[64], xb[64], rb_[64], rp[64], sh[64], sg[64], zb[64];
  const int n = blockIdx.x / Nn, b = blockIdx.x % Nn;
  const int d = threadIdx.x;
  const bool isS = d < 16;
  const int u = isS ? d : (d - 16) / 3;
  const int ic = isS ? 0 : (d - 16) % 3;

  xb[d] = x[((size_t)n * CHh + d) * Nn + b];
  float sumh = 0.0f, sumg = 0.0f;

  for (int a = 0; a < Nn; ++a) {
    __syncthreads();
    hc[d] = h_conv[(((size_t)n * CHh + d) * Nn + b) * Nn + a];
    xa[d] = x[((size_t)n * CHh + d) * Nn + a];
    __syncthreads();
    sumh += hc[d];
    float acc = 0.0f;                       // r1[n,d,a] + r2[n,d,b,a]
    if (isS) {
#pragma unroll
      for (int m = 0; m < 16; ++m)
        acc += r1w0[d * 16 + m] * xa[m] + r2w0[d * 16 + m] * hc[m];
    } else {
#pragma unroll
      for (int m = 0; m < 16; ++m)
        acc += r1w1[u * 16 + m] * xa[16 + 3 * m + ic] +
               r2w1[u * 16 + m] * hc[16 + 3 * m + ic];
    }
    __syncthreads();
    rb_[d] = acc;
    __syncthreads();
    float rv;                               // norm_act(., sigmoid)
    if (isS) rv = sigm(acc);
    else {
      const float v0 = rb_[16 + 3 * u], v1 = rb_[16 + 3 * u + 1], v2 = rb_[16 + 3 * u + 2];
      const float nn = sqrtf(v0 * v0 + v1 * v1 + v2 * v2 + EPSc);
      rv = acc * sigm(nn) / nn;
    }
    rp[d] = rv;
    __syncthreads();
    float fac;                              // gate_mul factor from r
    if (isS) fac = fabsf(rv);
    else {
      const float g0 = rp[16 + 3 * u], g1 = rp[16 + 3 * u + 1], g2 = rp[16 + 3 * u + 2];
      fac = sqrtf(g0 * g0 + g1 * g1 + g2 * g2 + EPSc);
    }
    sumg += fac * hc[d];
  }
  __syncthreads();
  sh[d] = sumh; sg[d] = sumg;
  __syncthreads();

  // z = norm_act(si(interleave(x_b, sum_h)), sigmoid)
  float za = 0.0f;
  if (isS) {
#pragma unroll
    for (int m = 0; m < 16; ++m)
      za += zw0[d * 32 + m] * xb[m] + zw0[d * 32 + 16 + m] * sh[m];
  } else {
#pragma unroll
    for (int m = 0; m < 16; ++m)
      za += zw1[u * 32 + m] * xb[16 + 3 * m + ic] +
            zw1[u * 32 + 16 + m] * sh[16 + 3 * m + ic];
  }
  rb_[d] = za;
  __syncthreads();
  float zv;
  if (isS) zv = sigm(za);
  else {
    const float v0 = rb_[16 + 3 * u], v1 = rb_[16 + 3 * u + 1], v2 = rb_[16 + 3 * u + 2];
    const float nn = sqrtf(v0 * v0 + v1 * v1 + v2 * v2 + EPSc);
    zv = za * sigm(nn) / nn;
  }
  zb[d] = zv;
  __syncthreads();

  // pre_h = norm_act(si(interleave(x_b, sum_g)), tanh)
  float pa = 0.0f;
  if (isS) {
#pragma unroll
    for (int m = 0; m < 16; ++m)
      pa += pw0[d * 32 + m] * xb[m] + pw0[d * 32 + 16 + m] * sg[m];
  } else {
#pragma unroll
    for (int m = 0; m < 16; ++m)
      pa += pw1[u * 32 + m] * xb[16 + 3 * m + ic] +
            pw1[u * 32 + 16 + m] * sg[16 + 3 * m + ic];
  }
  rb_[d] = pa;
  __syncthreads();
  float pv;
  if (isS) pv = tanhf(pa);
  else {
    const float v0 = rb_[16 + 3 * u], v1 = rb_[16 + 3 * u + 1], v2 = rb_[16 + 3 * u + 2];
    const float nn = sqrtf(v0 * v0 + v1 * v1 + v2 * v2 + EPSc);
    pv = pa * tanhf(nn) / nn;
  }
  float fac;                                // gate_mul factor from z
  if (isS) fac = fabsf(zv);
  else {
    const float g0 = zb[16 + 3 * u], g1 = zb[16 + 3 * u + 1], g2 = zb[16 + 3 * u + 2];
    fac = sqrtf(g0 * g0 + g1 * g1 + g2 * g2 + EPSc);
  }
  out[((size_t)n * CHh + d) * Nn + b] = sumh + fac * (pv + sumh);
}

extern "C" void kernel_launch(void* const* d_in, const int* in_sizes, int n_in,
                              void* d_out, int out_size, void* d_ws, size_t ws_size,
                              hipStream_t stream) {
  const float* x     = (const float*)d_in[0];
  const float* h_nei = (const float*)d_in[1];
  const float* hdiff = (const float*)d_in[2];
  const float* mask  = (const float*)d_in[3];
  const float* w00   = (const float*)d_in[4];
  const float* w10   = (const float*)d_in[5];
  const float* w01   = (const float*)d_in[6];
  const float* w11   = (const float*)d_in[7];
  const float* zw0   = (const float*)d_in[8];
  const float* zw1   = (const float*)d_in[9];
  const float* r1w0  = (const float*)d_in[10];
  const float* r1w1  = (const float*)d_in[11];
  const float* r2w0  = (const float*)d_in[12];
  const float* r2w1  = (const float*)d_in[13];
  const float* pw0   = (const float*)d_in[14];
  const float* pw1   = (const float*)d_in[15];

  float* h_conv = (float*)d_ws;   // B*CH*N*N f32 = 6.55 MB (L2-resident on MI455X)

  conv_kernel<<<Bb * Nn, 32, 0, stream>>>(h_nei, hdiff, mask, w00, w10, w01, w11, h_conv);
  post_kernel<<<Bb * Nn, 64, 0, stream>>>(x, h_conv, zw0, zw1, r1w0, r1w1, r2w0, r2w1,
                                          pw0, pw1, (float*)d_out);
}